// DifferentiableBoxParser_14542759264983
// MI455X (gfx1250) — compile-verified
//
#include <hip/hip_runtime.h>

typedef unsigned int u32;
typedef __attribute__((ext_vector_type(4))) unsigned int u32x4;
typedef __attribute__((ext_vector_type(4))) int i32x4;
typedef __attribute__((ext_vector_type(8))) int i32x8;

#define HO 256
#define WO 256
#define ELEMS (HO * WO)                    // 65536 elements per (b,k) slice
#define CHUNK_ROWS 32
#define CHUNK_ELEMS (CHUNK_ROWS * WO)      // 8192 elements = 32 KB per chunk
#define NCHUNK (ELEMS / CHUNK_ELEMS)       // 8 chunks
#define THREADS 512                        // 16 wave32s
#define CITERS (CHUNK_ELEMS / (THREADS * 4))  // 4 float4 per thread per chunk
#define NWAVES (THREADS / 32)

#if __has_builtin(__builtin_amdgcn_tensor_load_to_lds)
#define HAVE_TDM 1
#else
#define HAVE_TDM 0
#endif

__device__ __forceinline__ float wave_sum(float v) {
#pragma unroll
  for (int s = 16; s > 0; s >>= 1) v += __shfl_xor(v, s);
  return v;
}

#if HAVE_TDM
// Issue one TDM copy: 32 rows x 256 cols f32 from global -> LDS double buffer.
__device__ __forceinline__ void tdm_issue(const float* gbase, int chunk, int buf) {
  unsigned long long ga =
      (unsigned long long)(const void*)(gbase + (size_t)chunk * CHUNK_ELEMS);
  u32 lds_addr =
      __builtin_amdgcn_groupstaticsize() + (u32)buf * (CHUNK_ELEMS * 4u);

  // D# group 0: count=1 | lds_addr | global_addr[56:0] | type=2
  u32x4 g0;
  g0[0] = 1u;
  g0[1] = lds_addr;
  g0[2] = (u32)(ga & 0xFFFFFFFFull);
  g0[3] = (u32)((ga >> 32) & 0x01FFFFFFull) | (2u << 30);

  // D# group 1: data_size=4B, tensor 256x32, tile 256x32, dim0 stride 256
  i32x8 g1;
  g1[0] = 0x00020000;              // workgroup_mask=0, data_size=2 (4 bytes)
  g1[1] = (int)(WO << 16);         // tensor_dim0 = 256 (low 16 of 32)
  g1[2] = (int)(CHUNK_ROWS << 16); // tensor_dim0 hi=0, tensor_dim1 = 32 (low 16)
  g1[3] = (int)(WO << 16);         // tensor_dim1 hi=0, tile_dim0 = 256
  g1[4] = CHUNK_ROWS;              // tile_dim1 = 32, tile_dim2 = 0
  g1[5] = WO;                      // tensor_dim0_stride = 256
  g1[6] = 0;
  g1[7] = 0;

  i32x4 gz = (i32x4){0, 0, 0, 0};
#if __clang_major__ >= 23
  i32x8 gz8 = (i32x8){0, 0, 0, 0, 0, 0, 0, 0};
  __builtin_amdgcn_tensor_load_to_lds(g0, g1, gz, gz, gz8, 0);
#else
  __builtin_amdgcn_tensor_load_to_lds(g0, g1, gz, gz, 0);
#endif
}
#endif

__global__ __launch_bounds__(THREADS)
void dbp_kernel(const float* __restrict__ score,
                const float* __restrict__ offset,
                float* __restrict__ out) {
  extern __shared__ float lds[];          // 2 chunk buffers + reduction slots
  float* red = lds + 2 * CHUNK_ELEMS;

  const int bk   = blockIdx.x;            // b*8 + k
  const int tid  = threadIdx.x;
  const int lane = tid & 31;
  const int wave = tid >> 5;
  const float* slice = score + (size_t)bk * ELEMS;

  // Softmax is shift-invariant; with T=0.1 and |s| < ~6, exp2(s*C) stays well
  // inside f32 range, so no max-subtraction pass is needed (pure rescaling).
  const float C = 14.426950408889634f;    // log2(e) / TEMPERATURE

  float d = 0.f, sy = 0.f, sx = 0.f;

#if HAVE_TDM
  if (wave == 0) tdm_issue(slice, 0, 0);  // preload chunk 0 into buffer 0
#endif

  for (int c = 0; c < NCHUNK; ++c) {
#if HAVE_TDM
    if (wave == 0) __builtin_amdgcn_s_wait_tensorcnt(0);
    __syncthreads();                      // chunk c visible; buf (c+1)&1 free
    if (wave == 0 && c + 1 < NCHUNK) tdm_issue(slice, c + 1, (c + 1) & 1);
#else
    {
      const float4* src4 = (const float4*)(slice + (size_t)c * CHUNK_ELEMS);
      float4* dst4 = (float4*)(lds + (c & 1) * CHUNK_ELEMS);
#pragma unroll
      for (int it = 0; it < CITERS; ++it) {
        int j = it * THREADS + tid;
        dst4[j] = src4[j];
      }
      __syncthreads();
    }
#endif

    const float4* buf = (const float4*)(lds + (c & 1) * CHUNK_ELEMS);
    const int flatbase = c * CHUNK_ELEMS;
#pragma unroll
    for (int it = 0; it < CITERS; ++it) {
      int j = it * THREADS + tid;
      float4 v = buf[j];
      int flat = flatbase + j * 4;        // contiguous, never crosses a row
      float h  = (float)(flat >> 8);
      float w  = (float)(flat & 255);
      float e0 = exp2f(v.x * C);
      float e1 = exp2f(v.y * C);
      float e2 = exp2f(v.z * C);
      float e3 = exp2f(v.w * C);
      float es = e0 + e1 + e2 + e3;
      d  += es;
      sy += es * h;
      sx += e0 * w + e1 * (w + 1.f) + e2 * (w + 2.f) + e3 * (w + 3.f);
    }
#if HAVE_TDM
    __syncthreads();                      // all done with buf (c&1) before reuse
#endif
  }

  // ---------- Block reduction of (d, sy, sx) ----------
  d  = wave_sum(d);
  sy = wave_sum(sy);
  sx = wave_sum(sx);
  if (lane == 0) {
    red[wave] = d;
    red[NWAVES + wave] = sy;
    red[2 * NWAVES + wave] = sx;
  }
  __syncthreads();

  // ---------- Scalar epilogue: soft-ceil, smooth-clamp, gather, emit ----------
  if (tid == 0) {
    float D = 0.f, SY = 0.f, SX = 0.f;
    for (int i = 0; i < NWAVES; ++i) {
      D += red[i]; SY += red[NWAVES + i]; SX += red[2 * NWAVES + i];
    }
    float y = SY / D;
    float x = SX / D;

    // soft ceil: c + (1 - sigmoid(10*(c - floor(c))))
    float cy = y + (1.f - 1.f / (1.f + expf(-10.f * (y - floorf(y)))));
    float cx = x + (1.f - 1.f / (1.f + expf(-10.f * (x - floorf(x)))));

    // smooth clamp to [0, 255], SMOOTHNESS = 0.1 (sequential, like the reference)
    auto sclamp = [](float v, float lo, float hi) {
      if (v < lo) v = lo + 0.1f * tanhf((v - lo) * 10.f);
      if (v > hi) v = hi - 0.1f * tanhf((hi - v) * 10.f);
      return v;
    };
    float ycl = sclamp(cy, 0.f, (float)(HO - 1));
    float xcl = sclamp(cx, 0.f, (float)(WO - 1));

    int yi = (int)ycl;                    // truncation matches astype(int32)
    int xi = (int)xcl;

    const int b = bk >> 3;
    const int k = bk & 7;
    const float* op = offset + ((size_t)b * 16 + (size_t)(2 * k)) * (size_t)ELEMS;
    float oy = op[yi * WO + xi];
    float ox = op[ELEMS + yi * WO + xi];

    out[bk * 2 + 0] = (y + oy) * 4.f;     // STRIDE = 4
    out[bk * 2 + 1] = (x + ox) * 4.f;
  }
}

extern "C" void kernel_launch(void* const* d_in, const int* in_sizes, int n_in,
                              void* d_out, int out_size, void* d_ws, size_t ws_size,
                              hipStream_t stream) {
  const float* score  = (const float*)d_in[0];   // [64, 8, 256, 256] f32
  const float* offset = (const float*)d_in[1];   // [64, 16, 256, 256] f32
  float* out = (float*)d_out;                    // [64, 8, 2] f32

  const int bk = in_sizes[0] / ELEMS;            // 512 slices
  const size_t shmem = (size_t)(2 * CHUNK_ELEMS + 64) * sizeof(float);  // ~64.3 KB

  dbp_kernel<<<dim3(bk), dim3(THREADS), shmem, stream>>>(score, offset, out);
}